// MatrixProductState_42107859370890
// MI455X (gfx1250) — compile-verified
//
#include <hip/hip_runtime.h>

typedef float v2f __attribute__((ext_vector_type(2)));
typedef float v4f __attribute__((ext_vector_type(4)));
typedef float v8f __attribute__((ext_vector_type(8)));

#define BATCH   16384
#define IDIM    2048
#define NSTEPS  1023      // sites 1..1023 (site 0 handled at init)
#define SITE_F  2048      // floats per site tensor (32*2*32)
#define BLOCKT  128       // 4 waves
#define BPB     64        // batch rows per block (4 waves * 16)

// swap lane halves (lane ^ 16) within wave32: ds_swizzle SWAPX16
__device__ __forceinline__ float swz16(float v) {
  return __int_as_float(__builtin_amdgcn_ds_swizzle(__float_as_int(v), 0x401f));
}

__global__ __launch_bounds__(BLOCKT) void mps_wmma_kernel(
    const float* __restrict__ x,
    const float* __restrict__ A_first,
    const float* __restrict__ A_mid,
    const float* __restrict__ A_last,
    float* __restrict__ out)
{
  __shared__ float smem[2 * SITE_F];   // double-buffered A-fragments, 16 KB

  const int tid  = threadIdx.x;
  const int lane = tid & 31;
  const int col  = lane & 15;          // batch column within the 16-wide tile
  const int half = lane >> 4;          // lane half (K split / M split)
  const int wv   = tid >> 5;
  const int b    = blockIdx.x * BPB + wv * 16 + col;

  // ---- staging constants: thread tid fills fragment f = (p*2+t)*8+k,
  //      pair-entries lane_e = le0 .. le0+7 (value = A[c][p][r] with
  //      c = 4k + 2*(lane_e/16) + j, r = 16t + lane_e%16)
  const int f    = tid >> 2;
  const int le0  = (tid & 3) << 3;             // 0,8,16,24
  const int sp   = f >> 4;
  const int st   = (f >> 3) & 1;
  const int sk   = f & 7;
  const int c0   = 4 * sk + ((le0 >> 4) << 1);
  const int rr0  = 16 * st + (le0 & 15);
  const int goff = (c0 * 2 + sp) * 32 + rr0;   // float offset into site tensor
  float* ldst0 = smem + tid * 16;              // 64 B per thread per buffer
  float* ldst1 = smem + SITE_F + tid * 16;

  const v2f* xrow = (const v2f*)(x + (size_t)b * IDIM);

  // ---- site 0: R^T[r][b] = A_first[0][r]*x0 + A_first[1][r]*x1, built
  //      directly in WMMA C/D layout (VGPR g holds row g + 8*half of tile)
  v8f r0, r1;
  {
    v2f xv0 = xrow[0];
    #pragma unroll
    for (int g = 0; g < 8; ++g) {
      const int rA = g + 8 * half;
      r0[g] = A_first[rA]      * xv0.x + A_first[32 + rA] * xv0.y;
      r1[g] = A_first[16 + rA] * xv0.x + A_first[48 + rA] * xv0.y;
    }
  }

  // ---- prefetch + stage step 0 (site 1 = A_mid[0]) into buffer 0
  v4f sA0, sA1, sB0, sB1;
  {
    const v4f* g0 = (const v4f*)(A_mid + goff);
    const v4f* g1 = (const v4f*)(A_mid + goff + 64);   // row c0+1 (+256 B)
    sA0 = g0[0]; sA1 = g0[1];
    sB0 = g1[0]; sB1 = g1[1];
    v4f* d = (v4f*)ldst0;
    v4f p0 = { sA0.x, sB0.x, sA0.y, sB0.y };
    v4f p1 = { sA0.z, sB0.z, sA0.w, sB0.w };
    v4f p2 = { sA1.x, sB1.x, sA1.y, sB1.y };
    v4f p3 = { sA1.z, sB1.z, sA1.w, sB1.w };
    d[0] = p0; d[1] = p1; d[2] = p2; d[3] = p3;
  }
  v2f xv = xrow[1];

  for (int step = 0; step < NSTEPS; ++step) {
    __syncthreads();   // fragments for this step visible; prev buffer free

    const bool more = (step + 1) < NSTEPS;
    v2f xn = xv;
    if (more) {        // uniform branch: prefetch next site (A + x) to regs
      const float* As = ((step + 1) < (NSTEPS - 1))
                          ? (A_mid + (size_t)(step + 1) * SITE_F)
                          : A_last;
      const v4f* g0 = (const v4f*)(As + goff);
      const v4f* g1 = (const v4f*)(As + goff + 64);
      sA0 = g0[0]; sA1 = g0[1];
      sB0 = g1[0]; sB1 = g1[1];
      xn = xrow[step + 2];
    }

    const v2f* frag = (const v2f*)(smem + (step & 1) * SITE_F);

    // half-swapped copies of the state for B-operand assembly
    v8f w0, w1;
    #pragma unroll
    for (int g = 0; g < 8; ++g) { w0[g] = swz16(r0[g]); w1[g] = swz16(r1[g]); }

    v8f aP0T0 = {0,0,0,0,0,0,0,0};
    v8f aP1T0 = {0,0,0,0,0,0,0,0};
    v8f aP0T1 = {0,0,0,0,0,0,0,0};
    v8f aP1T1 = {0,0,0,0,0,0,0,0};

    #pragma unroll
    for (int k = 0; k < 8; ++k) {       // K chunks of 4 over BOND=32
      const int bo  = (k & 1) * 4;
      const int dir = (k >> 1) & 1;
      float dxlo, dxhi, dylo, dyhi;
      if (k < 4) {
        dxlo = dir ? w0[bo]     : r0[bo];
        dxhi = dir ? r0[bo + 2] : w0[bo + 2];
        dylo = dir ? w0[bo + 1] : r0[bo + 1];
        dyhi = dir ? r0[bo + 3] : w0[bo + 3];
      } else {
        dxlo = dir ? w1[bo]     : r1[bo];
        dxhi = dir ? r1[bo + 2] : w1[bo + 2];
        dylo = dir ? w1[bo + 1] : r1[bo + 1];
        dyhi = dir ? r1[bo + 3] : w1[bo + 3];
      }
      const float bx = half ? dxhi : dxlo;   // B[4k+2*half  ][col]
      const float by = half ? dyhi : dylo;   // B[4k+2*half+1][col]
      v2f bP0 = { bx * xv.x, by * xv.x };    // B fragment scaled by x0[b]
      v2f bP1 = { bx * xv.y, by * xv.y };    // B fragment scaled by x1[b]

      v2f a00 = frag[(0 * 8 + k) * 32 + lane];   // A0^T rows  0..15
      v2f a01 = frag[(1 * 8 + k) * 32 + lane];   // A0^T rows 16..31
      v2f a10 = frag[(2 * 8 + k) * 32 + lane];   // A1^T rows  0..15
      v2f a11 = frag[(3 * 8 + k) * 32 + lane];   // A1^T rows 16..31

      aP0T0 = __builtin_amdgcn_wmma_f32_16x16x4_f32(false, a00, false, bP0, (short)0, aP0T0, false, false);
      aP1T0 = __builtin_amdgcn_wmma_f32_16x16x4_f32(false, a10, false, bP1, (short)0, aP1T0, false, false);
      aP0T1 = __builtin_amdgcn_wmma_f32_16x16x4_f32(false, a01, false, bP0, (short)0, aP0T1, false, false);
      aP1T1 = __builtin_amdgcn_wmma_f32_16x16x4_f32(false, a11, false, bP1, (short)0, aP1T1, false, false);
    }
    r0 = aP0T0 + aP1T0;
    r1 = aP0T1 + aP1T1;
    xv = xn;

    if (more) {       // stage next site's fragments into the other buffer
      v4f* d = (v4f*)(((step + 1) & 1) ? ldst1 : ldst0);
      v4f p0 = { sA0.x, sB0.x, sA0.y, sB0.y };
      v4f p1 = { sA0.z, sB0.z, sA0.w, sB0.w };
      v4f p2 = { sA1.x, sB1.x, sA1.y, sB1.y };
      v4f p3 = { sA1.z, sB1.z, sA1.w, sB1.w };
      d[0] = p0; d[1] = p1; d[2] = p2; d[3] = p3;
    }
  }

  // ---- write out: out[b][o] = R^T_final[o][b]
  float* orow = out + (size_t)b * 32;
  #pragma unroll
  for (int g = 0; g < 8; ++g) {
    orow[g + 8 * half]      = r0[g];
    orow[16 + g + 8 * half] = r1[g];
  }
}

extern "C" void kernel_launch(void* const* d_in, const int* in_sizes, int n_in,
                              void* d_out, int out_size, void* d_ws, size_t ws_size,
                              hipStream_t stream) {
  const float* x  = (const float*)d_in[0];   // (16384, 2048) f32
  const float* Af = (const float*)d_in[1];   // (1, 2, 32)    f32
  const float* Am = (const float*)d_in[2];   // (1022,32,2,32) f32
  const float* Al = (const float*)d_in[3];   // (32, 2, 32)   f32
  float* o = (float*)d_out;                  // (16384, 32)   f32
  (void)in_sizes; (void)n_in; (void)out_size; (void)d_ws; (void)ws_size;
  mps_wmma_kernel<<<dim3(BATCH / BPB), dim3(BLOCKT), 0, stream>>>(x, Af, Am, Al, o);
}